// gGASingleOrb_26061861552601
// MI455X (gfx1250) — compile-verified
//
#include <hip/hip_runtime.h>
#include <hip/hip_bf16.h>
#include <math.h>

// ---------------------------------------------------------------------------
// GA single-orbital embedding kernel for MI455X (gfx1250, wave32, WMMA).
//
// Heavy math = three tiny-M huge-K fp32 GEMMs (16x16, K up to 1M) expressed
// with V_WMMA_F32_16X16X4_F32 to keep fp32 accumulation precision.
// ---------------------------------------------------------------------------

#define GA_DA   65536      // 4^8 aux Fock dim
#define GA_DP   16         // 4^2 physical Fock dim
#define GA_NSO  16         // 2 * N_AUX_ORB aux spin-orbitals

typedef __attribute__((ext_vector_type(2))) float v2f;
typedef __attribute__((ext_vector_type(8))) float v8f;

#define WMMA_F32_K4(a, b, c) \
  __builtin_amdgcn_wmma_f32_16x16x4_f32(false, (a), false, (b), (short)0, (c), false, false)

// ---------------------------------------------------------------------------
// ws layout (floats):
//   [0 .. 1048575]      sbT : transposed sb, sbT[j*16 + i] = sb[i, j]
//   [1048576 .. +255]   accS  (sb @ sb^T, 16x16)
//   [1048832 .. +255]   accRD (16x16)
//   [1049088 .. +255]   accB  (16x16, only rows 0..3 meaningful)
//   [1049344]           normsq
// ---------------------------------------------------------------------------
#define WS_SBT   0
#define WS_ACCS  1048576
#define WS_ACCRD (WS_ACCS + 256)
#define WS_ACCB  (WS_ACCRD + 256)
#define WS_NRM   (WS_ACCB + 256)
#define WS_ZERO_N (WS_NRM + 1)

__global__ __launch_bounds__(256) void ga_zero(float* __restrict__ ws, int n) {
  int i = blockIdx.x * blockDim.x + threadIdx.x;
  if (i < n) ws[i] = 0.0f;
}

__global__ __launch_bounds__(256) void ga_norm(const float* __restrict__ x, int n,
                                               float* __restrict__ normsq) {
  __shared__ float red[256];
  int t = threadIdx.x;
  float s = 0.0f;
  for (int i = blockIdx.x * 256 + t; i < n; i += gridDim.x * 256) {
    float v = x[i];
    s += v * v;
  }
  red[t] = s;
  __syncthreads();
  for (int st = 128; st > 0; st >>= 1) {
    if (t < st) red[t] += red[t + st];
    __syncthreads();
  }
  if (t == 0) atomicAdd(normsq, red[0]);
}

__global__ __launch_bounds__(256) void ga_scatter(const float* __restrict__ amp,
                                                  const int* __restrict__ iphy,
                                                  const int* __restrict__ iaux, int n,
                                                  const float* __restrict__ normsq,
                                                  float* __restrict__ sbT) {
  int i = blockIdx.x * blockDim.x + threadIdx.x;
  if (i < n) {
    float rn = rsqrtf(*normsq);
    sbT[(size_t)iaux[i] * 16 + iphy[i]] = amp[i] * rn;
  }
}

// ---------------------------------------------------------------------------
// Main WMMA kernel: 128 blocks x 256 threads = 1024 waves, 64 aux states each.
// Per aux state j:
//   lane p = lane%16 owns row p of
//     G[p,i] = coefM[p,j] * sbT[partner_a[p,j]*16 + i]   (RD Gram)
//     N[p,i] = coefN[p,j] * sbT[partner_a[p,j]*16 + i]
//     L[q,i] = coefL_p[q,i] * sbT[j*16 + partner_p[q,i]] (q<4, else 0)
//   RD += G G^T (4x WMMA K=4), B += L N^T (4x WMMA K=4)
// Per 4 states: S += colJ colJ^T via one WMMA (A[m,k] = sbT[(j0+k)*16+m]).
// ---------------------------------------------------------------------------
__global__ __launch_bounds__(256) void ga_main(const float* __restrict__ sbT,
                                               const float* __restrict__ coefM,
                                               const float* __restrict__ coefN,
                                               const int* __restrict__ partnerA,
                                               const float* __restrict__ coefL,
                                               const int* __restrict__ partnerP,
                                               float* __restrict__ accS,
                                               float* __restrict__ accRD,
                                               float* __restrict__ accB) {
  __shared__ float sLp[256];  // coefL_p padded to 16x16 (rows >=4 zero)
  __shared__ int   sPp[256];  // partner_p padded to 16x16
  __shared__ float redS[256], redRD[256], redB[256];

  int t = threadIdx.x;
  {
    int q = t >> 4, i = t & 15;
    sLp[t] = (q < 4) ? coefL[q * 16 + i] : 0.0f;
    sPp[t] = (q < 4) ? partnerP[q * 16 + i] : 0;
    redS[t] = 0.0f;
    redRD[t] = 0.0f;
    redB[t] = 0.0f;
  }
  __syncthreads();

  const int lane = t & 31;
  const int half = lane >> 4;  // selects K pair {2*half, 2*half+1} in each K=4 chunk
  const int m    = lane & 15;  // row (A) / col (B,C,D) index

  const int gwave = blockIdx.x * 8 + (t >> 5);
  const int J_PER_WAVE = GA_DA / (128 * 8);  // 64
  const int jstart = gwave * J_PER_WAVE;

  v8f aS = {};
  v8f aRD = {};
  v8f aB = {};

  for (int j0 = jstart; j0 < jstart + J_PER_WAVE; j0 += 4) {
    // --- S (Gram of sb columns), 4 states per WMMA ---
    v2f sA;
    sA.x = sbT[(size_t)(j0 + 2 * half    ) * 16 + m];
    sA.y = sbT[(size_t)(j0 + 2 * half + 1) * 16 + m];
    aS = WMMA_F32_K4(sA, sA, aS);

#pragma unroll
    for (int jj = 0; jj < 4; ++jj) {
      const int j = j0 + jj;
      const int   idx = partnerA[m * GA_DA + j];
      const float mv  = coefM[m * GA_DA + j];
      const float nv  = coefN[m * GA_DA + j];
      // gathered column base, pre-offset by this lane's K-pair position
      const float* colP = sbT + (size_t)idx * 16 + 2 * half;
      const float* colJ = sbT + (size_t)j * 16;

#pragma unroll
      for (int kk = 0; kk < 4; ++kk) {
        const float2 cp = *(const float2*)(colP + kk * 4);  // 8B-aligned
        v2f aG;
        aG.x = mv * cp.x;
        aG.y = mv * cp.y;
        aRD = WMMA_F32_K4(aG, aG, aRD);  // Gram: A-frag == B-frag

        v2f bN;
        bN.x = nv * cp.x;
        bN.y = nv * cp.y;
        const int k0 = kk * 4 + 2 * half;
        v2f aL;
        aL.x = sLp[m * 16 + k0    ] * colJ[sPp[m * 16 + k0    ]];
        aL.y = sLp[m * 16 + k0 + 1] * colJ[sPp[m * 16 + k0 + 1]];
        aB = WMMA_F32_K4(aL, bN, aB);
      }
    }
  }

  // C/D fragment layout: VGPR r holds (M = r + 8*half, N = m)
#pragma unroll
  for (int r = 0; r < 8; ++r) {
    const int row = r + 8 * half;
    atomicAdd(&redS[row * 16 + m],  aS[r]);
    atomicAdd(&redRD[row * 16 + m], aRD[r]);
    atomicAdd(&redB[row * 16 + m],  aB[r]);
  }
  __syncthreads();
  atomicAdd(&accS[t],  redS[t]);
  atomicAdd(&accRD[t], redRD[t]);
  atomicAdd(&accB[t],  redB[t]);
}

// ---------------------------------------------------------------------------
// Finalize: U, RD output, C = RD - RD^2, Jacobi eigensolve (symmetric symsqrt),
// A = V sqrt(|w|) V^T, solve A X = B^T, write R. All tiny (16x16) serial work.
// ---------------------------------------------------------------------------
__global__ __launch_bounds__(256) void ga_finalize(const float* __restrict__ accS,
                                                   const float* __restrict__ accRD,
                                                   const float* __restrict__ accB,
                                                   const float* __restrict__ Hint,
                                                   float* __restrict__ out) {
  int t = threadIdx.x;
  out[1 + t] = accRD[t];  // RD.reshape(8,2,8,2) == flat row-major 16x16

  if (t == 0) {
    // U = sum(Hint * S)
    float U = 0.0f;
    for (int i = 0; i < 256; ++i) U += Hint[i] * accS[i];
    out[0] = U;

    float RD[16][16], C[16][16], V[16][16];
    for (int i = 0; i < 16; ++i)
      for (int j = 0; j < 16; ++j) RD[i][j] = accRD[i * 16 + j];
    // C = RD (I - RD) = RD - RD^2  (symmetric)
    for (int i = 0; i < 16; ++i)
      for (int j = 0; j < 16; ++j) {
        float s = 0.0f;
        for (int k = 0; k < 16; ++k) s += RD[i][k] * RD[k][j];
        C[i][j] = RD[i][j] - s;
      }
    // Jacobi eigendecomposition
    for (int i = 0; i < 16; ++i)
      for (int j = 0; j < 16; ++j) V[i][j] = (i == j) ? 1.0f : 0.0f;
    for (int sweep = 0; sweep < 30; ++sweep) {
      float off = 0.0f;
      for (int p = 0; p < 15; ++p)
        for (int q = p + 1; q < 16; ++q) off += C[p][q] * C[p][q];
      if (off < 1e-24f) break;
      for (int p = 0; p < 15; ++p)
        for (int q = p + 1; q < 16; ++q) {
          const float apq = C[p][q];
          if (fabsf(apq) < 1e-30f) continue;
          const float app = C[p][p], aqq = C[q][q];
          const float tau = (aqq - app) / (2.0f * apq);
          const float tt =
              (tau >= 0.0f ? 1.0f : -1.0f) / (fabsf(tau) + sqrtf(1.0f + tau * tau));
          const float c = 1.0f / sqrtf(1.0f + tt * tt);
          const float s = tt * c;
          for (int k = 0; k < 16; ++k) {
            const float ckp = C[k][p], ckq = C[k][q];
            C[k][p] = c * ckp - s * ckq;
            C[k][q] = s * ckp + c * ckq;
          }
          for (int k = 0; k < 16; ++k) {
            const float cpk = C[p][k], cqk = C[q][k];
            C[p][k] = c * cpk - s * cqk;
            C[q][k] = s * cpk + c * cqk;
          }
          for (int k = 0; k < 16; ++k) {
            const float vkp = V[k][p], vkq = V[k][q];
            V[k][p] = c * vkp - s * vkq;
            V[k][q] = s * vkp + c * vkq;
          }
        }
    }
    // symsqrt: singular values = |eigenvalues|; threshold as in reference
    float w[16], smax = 0.0f;
    for (int k = 0; k < 16; ++k) {
      w[k] = C[k][k];
      smax = fmaxf(smax, fabsf(w[k]));
    }
    const float thresh = smax * 16.0f * 1.1920929e-7f;
    float rs[16];
    for (int k = 0; k < 16; ++k) {
      const float a = fabsf(w[k]);
      rs[k] = (a > thresh) ? sqrtf(a) : 0.0f;
    }
    // A = V diag(rs) V^T (symmetric, so A.T == A); solve A X = B^T (16x4)
    float Aug[16][20];
    for (int i = 0; i < 16; ++i) {
      for (int j = 0; j < 16; ++j) {
        float s2 = 0.0f;
        for (int k = 0; k < 16; ++k) s2 += rs[k] * V[i][k] * V[j][k];
        Aug[i][j] = s2;
      }
      for (int q = 0; q < 4; ++q) Aug[i][16 + q] = accB[q * 16 + i];
    }
    // Gauss-Jordan with partial pivoting
    for (int col = 0; col < 16; ++col) {
      int piv = col;
      float best = fabsf(Aug[col][col]);
      for (int r = col + 1; r < 16; ++r) {
        const float a = fabsf(Aug[r][col]);
        if (a > best) { best = a; piv = r; }
      }
      if (piv != col)
        for (int k = 0; k < 20; ++k) {
          const float tmp = Aug[col][k];
          Aug[col][k] = Aug[piv][k];
          Aug[piv][k] = tmp;
        }
      const float d = Aug[col][col];
      const float invd = (fabsf(d) > 1e-30f) ? 1.0f / d : 0.0f;
      for (int k = col; k < 20; ++k) Aug[col][k] *= invd;
      for (int r = 0; r < 16; ++r) {
        if (r == col) continue;
        const float f = Aug[r][col];
        if (f != 0.0f)
          for (int k = col; k < 20; ++k) Aug[r][k] -= f * Aug[col][k];
      }
    }
    for (int p = 0; p < 16; ++p)
      for (int q = 0; q < 4; ++q) out[257 + p * 4 + q] = Aug[p][16 + q];
  }
}

extern "C" void kernel_launch(void* const* d_in, const int* in_sizes, int n_in,
                              void* d_out, int out_size, void* d_ws, size_t ws_size,
                              hipStream_t stream) {
  const float* SBamp    = (const float*)d_in[0];
  const float* Hint     = (const float*)d_in[1];
  const float* coefM_a  = (const float*)d_in[2];
  const float* coefN_a  = (const float*)d_in[3];
  const float* coefL_p  = (const float*)d_in[4];
  const int*   partnerA = (const int*)d_in[5];
  const int*   partnerP = (const int*)d_in[6];
  const int*   idx_phy  = (const int*)d_in[7];
  const int*   idx_aux  = (const int*)d_in[8];
  float* out = (float*)d_out;
  float* ws  = (float*)d_ws;

  float* sbT   = ws + WS_SBT;
  float* accS  = ws + WS_ACCS;
  float* accRD = ws + WS_ACCRD;
  float* accB  = ws + WS_ACCB;
  float* nrm   = ws + WS_NRM;

  const int n = in_sizes[0];

  ga_zero<<<(WS_ZERO_N + 255) / 256, 256, 0, stream>>>(ws, WS_ZERO_N);
  ga_norm<<<256, 256, 0, stream>>>(SBamp, n, nrm);
  ga_scatter<<<(n + 255) / 256, 256, 0, stream>>>(SBamp, idx_phy, idx_aux, n, nrm, sbT);
  ga_main<<<128, 256, 0, stream>>>(sbT, coefM_a, coefN_a, partnerA, coefL_p, partnerP,
                                   accS, accRD, accB);
  ga_finalize<<<1, 256, 0, stream>>>(accS, accRD, accB, Hint, out);
  (void)n_in; (void)out_size; (void)ws_size;
}